// LocalDiscriminator_73169062855279
// MI455X (gfx1250) — compile-verified
//
#include <hip/hip_runtime.h>
#include <hip/hip_bf16.h>

// LocalDiscriminator: logits = einsum('bchw,c->bhw', x, w) + b; z = sigmoid(logits);
// loss = mean(softplus(z) - z*mode).  B=32, C=1024, H=W=64.
// Memory-bound (512 MB streamed once). Channel-dot mapped onto V_WMMA_F32_16X16X4_F32:
//   A (16x4)  = w[c0..c0+3] broadcast to every row (from LDS, float2 per lane)
//   B (4x16)  = x tile: column n = spatial s0+n, rows = 4 consecutive channels
//   D (16x16) accumulates; all rows identical => lane l holds logits[s0 + l%16].

typedef float v2f __attribute__((ext_vector_type(2)));
typedef float v8f __attribute__((ext_vector_type(8)));

#define NB   32
#define NC   1024
#define NHW  4096            // 64*64
#define NTOT (NB * NHW)      // 131072 outputs
#define TILES (NTOT / 16)    // 8192 wave-tiles
#define WAVES_PER_BLOCK 8
#define NBLOCKS (TILES / WAVES_PER_BLOCK)  // 1024

__global__ __launch_bounds__(256)
void disc_stage1(const float* __restrict__ x,
                 const float* __restrict__ w,
                 const float* __restrict__ bias,
                 const int*   __restrict__ mode,
                 float*       __restrict__ partial)
{
    __shared__ float wl[NC];
    // Stage the 4 KB weight vector into LDS once per block (uniform loop, no divergence).
    for (int i = threadIdx.x; i < NC; i += 256) wl[i] = w[i];
    __syncthreads();

    const int lane = threadIdx.x & 31;
    const int wave = threadIdx.x >> 5;
    const int tile = blockIdx.x * WAVES_PER_BLOCK + wave;   // 0..8191
    const int batch = tile >> 8;                            // 256 tiles per batch image
    const int s0    = (tile & 255) << 4;                    // 16 consecutive pixels
    const int l16   = lane & 15;
    const int hi    = lane >> 4;                            // 0: K slots {0,1}, 1: K slots {2,3}

    // Per-lane x pointer: channel (2*hi), pixel s0+l16.  Per iteration we read this
    // channel and the next one (+NHW floats), then jump 4 channels (+4*NHW).
    const float* px = x + ((size_t)batch * NC + 2 * hi) * NHW + (s0 + l16);

    v8f acc = {};   // f32 16x16 accumulator (8 VGPRs)
#pragma unroll 4
    for (int c0 = 0; c0 < NC; c0 += 4) {
        // A: w[c0+2*hi] -> K slot (2*hi) in v0, w[c0+2*hi+1] -> K slot (2*hi+1) in v1.
        const float2 wv = *(const float2*)(&wl[c0 + 2 * hi]);
        v2f av; av.x = wv.x; av.y = wv.y;
        // B: x[c0+2*hi][s0+l16] in v0, x[c0+2*hi+1][s0+l16] in v1 (mirrors A's K layout).
        v2f bv; bv.x = px[0]; bv.y = px[NHW];
        acc = __builtin_amdgcn_wmma_f32_16x16x4_f32(
            /*neg_a=*/false, av, /*neg_b=*/false, bv,
            /*c_mod=*/(short)0, acc, /*reuse_a=*/false, /*reuse_b=*/false);
        px += 4 * NHW;
    }

    // Every D VGPR of lane l holds logits[s0 + (l & 15)]; lanes 16..31 are duplicates.
    float logit = acc[0] + bias[0];
    float z  = 1.0f / (1.0f + __expf(-logit));
    float y  = (float)mode[0];
    float pe = log1pf(__expf(z)) - z * y;      // softplus(z) - z*y, z in (0,1): stable
    if (hi) pe = 0.0f;                          // drop duplicate half-wave

    // Fixed-order wave reduction (deterministic).
    for (int off = 16; off >= 1; off >>= 1)
        pe += __shfl_xor(pe, off, 32);

    if (lane == 0) partial[tile] = pe;
}

__global__ __launch_bounds__(256)
void disc_stage2(const float* __restrict__ partial, float* __restrict__ out)
{
    __shared__ float red[256];
    float s = 0.0f;
    // Fixed traversal order -> deterministic sum of the 8192 wave partials.
    for (int j = 0; j < TILES / 256; ++j)
        s += partial[threadIdx.x + 256 * j];
    red[threadIdx.x] = s;
    __syncthreads();
    for (int stride = 128; stride >= 1; stride >>= 1) {
        if ((int)threadIdx.x < stride) red[threadIdx.x] += red[threadIdx.x + stride];
        __syncthreads();
    }
    if (threadIdx.x == 0) out[0] = red[0] * (1.0f / (float)NTOT);
}

extern "C" void kernel_launch(void* const* d_in, const int* in_sizes, int n_in,
                              void* d_out, int out_size, void* d_ws, size_t ws_size,
                              hipStream_t stream)
{
    const float* x    = (const float*)d_in[0];   // [32,1024,64,64] f32
    const float* w    = (const float*)d_in[1];   // [1,1024] f32
    const float* b    = (const float*)d_in[2];   // [1] f32
    const int*   mode = (const int*)d_in[3];     // scalar int
    float* partial = (float*)d_ws;               // 8192 floats = 32 KB scratch
    float* out     = (float*)d_out;              // 1 float

    disc_stage1<<<NBLOCKS, 256, 0, stream>>>(x, w, b, mode, partial);
    disc_stage2<<<1, 256, 0, stream>>>(partial, out);
}